// SpikingPineal_48842368090342
// MI455X (gfx1250) — compile-verified
//
#include <hip/hip_runtime.h>

typedef __attribute__((ext_vector_type(2))) float v2f;
typedef __attribute__((ext_vector_type(8))) float v8f;

// Single-wave latency-optimized kernel for the SpikingPineal reference.
// Lanes 0-5: Izhikevich neurons. Lanes 0-7: predictive-coding cells.
// Tail reductions via one V_WMMA_F32_16X16X4_F32 (row-sums against a
// ones B-matrix; exact in f32).
__launch_bounds__(32)
__global__ void SpikingPineal_kernel(const float* __restrict__ light_level,
                                     const float* __restrict__ retinal_luminance,
                                     const float* __restrict__ melatonin_raw,
                                     const float* __restrict__ mel_ema,
                                     const float* __restrict__ v_in,
                                     const float* __restrict__ u_in,
                                     const float* __restrict__ rate_in,
                                     const float* __restrict__ noise,     // [10][6]
                                     const float* __restrict__ v_soma,    // [2][4]
                                     const float* __restrict__ v_apical,  // [2][4]
                                     const float* __restrict__ precision, // [2][4]
                                     float* __restrict__ out)             // [7]
{
    __shared__ float buf[64];
    const int lane = threadIdx.x;

    // Zero the staging buffer: rows 8-15 of the A-matrix and the rate
    // padding slots (buf[30], buf[31]) must read as 0.
    buf[lane]      = 0.0f;
    buf[lane + 32] = 0.0f;
    __syncthreads();

    // ---- light / melatonin chemistry (uniform across all lanes) ----
    const float light   = fminf(1.0f, light_level[0] * 0.6f + retinal_luminance[0] * 0.4f);
    const float mraw    = melatonin_raw[0];
    const float mel_sup = fmaxf(0.0f, mraw - 0.1f * light);               // SUPPRESSION_RATE
    const float mel_syn = fminf(1.0f, mraw + 0.05f * (1.0f - light));     // SYNTHESIS_RATE
    const float mel     = (light > 0.3f) ? mel_sup : mel_syn;
    const float ema     = mel_ema[0] + 0.05f * (mel - mel_ema[0]);        // EMA_ALPHA

    // ---- Izhikevich neurons: lanes 0..5, 10 fully-unrolled steps ----
    if (lane < 6) {
        float Ibase;
        switch (lane) {
        case 0:  Ibase = light * 15.0f;          break;
        case 1:  Ibase = light * 10.0f;          break;
        case 2:  Ibase = (1.0f - light) * 8.0f;  break;
        case 3:  Ibase = mel * 12.0f;            break;
        case 4:  Ibase = mel * 8.0f;             break;
        default: Ibase = (1.0f - mel) * 6.0f;    break;
        }
        float v = v_in[lane], u = u_in[lane], rate = rate_in[lane];
        #pragma unroll
        for (int t = 0; t < 10; ++t) {
            const float Iin = Ibase - 1.0f + noise[t * 6 + lane] * 0.3f;  // I_TONIC
            const float v2  = v + (0.04f * v * v + 5.0f * v + 140.0f - u + Iin);
            const float u2  = u + 0.02f * (0.2f * v - u);                 // A, B
            const float spk = (v2 >= 30.0f) ? 1.0f : 0.0f;
            v = (spk > 0.0f) ? -65.0f : v2;                               // C
            u = u2 + 8.0f * spk;                                          // D
            rate = 0.95f * rate + 0.05f * spk;                            // RATE_DECAY
        }
        buf[24 + lane] = rate;                                            // rate[0..5]
    }

    // ---- two-compartment PC column: lanes 0..7, 8 substeps ----
    if (lane < 8) {
        const int   ch = lane >> 2;
        const float s  = ch ? mel : light;   // sensory
        const float p  = ch ? ema : light;   // prediction
        float vs = v_soma[lane], va = v_apical[lane];
        #pragma unroll
        for (int t = 0; t < 8; ++t) {
            const float vs2 = vs + 0.1f * (-vs + s + 0.5f * tanhf(va));   // PC_DT
            const float va2 = va + 0.1f * (-va + p);
            vs = vs2; va = va2;
        }
        const float pe   = s - tanhf(vs);
        const float prec = precision[lane] + 0.1f * (1.0f / (1.0f + pe * pe) - precision[lane]);
        buf[lane]        = fabsf(pe);        // rows 0-1: |pe|
        buf[8  + lane]   = prec;             // rows 2-3: precision
        buf[16 + lane]   = prec * pe * pe;   // rows 4-5: free-energy terms
    }
    __syncthreads();

    // ---- tail reductions: one V_WMMA_F32_16X16X4_F32 row-sum ----
    // A (16x4 f32) per ISA layout: lanes 0-15 hold {K=0,K=1}, lanes 16-31 hold
    // {K=2,K=3} of row M = lane%16.  A[r][k] = buf[4r+k]; rows 8-15 are zero.
    const int base = 4 * (lane & 15) + 2 * (lane >> 4);
    v2f a; a.x = buf[base]; a.y = buf[base + 1];
    v2f b; b.x = 1.0f;      b.y = 1.0f;        // B = ones (layout-independent)
    v8f c = {};

#if __has_builtin(__builtin_amdgcn_wmma_f32_16x16x4_f32)
    // 8 args: (neg_a, A, neg_b, B, c_mod, C, reuse_a, reuse_b)
    const v8f d = __builtin_amdgcn_wmma_f32_16x16x4_f32(
        false, a, false, b, (short)0, c, false, false);
    // Lane 0, VGPR j = D[M=j][N=0] = sum_k A[j][k]  (C/D layout, lanes 0-15).
    if (lane == 0) {
        out[0] = light;
        out[1] = mel;
        out[2] = ema;
        out[3] = (d[0] + d[1]) * (1.0f / 8.0f);   // mean |pe|
        out[4] = (d[2] + d[3]) * (1.0f / 8.0f);   // mean precision
        out[5] = 0.5f * (d[4] + d[5]);            // free energy
        out[6] = (d[6] + d[7]) * (1.0f / 6.0f);   // mean rate
    }
#else
    // Scalar fallback (keeps the kernel compiling if the builtin
    // is missing on this toolchain).
    (void)a; (void)b; (void)c;
    if (lane == 0) {
        float pe_s = 0.f, pr_s = 0.f, fe_s = 0.f, rt_s = 0.f;
        for (int i = 0; i < 8; ++i) { pe_s += buf[i]; pr_s += buf[8 + i]; fe_s += buf[16 + i]; }
        for (int i = 0; i < 6; ++i) { rt_s += buf[24 + i]; }
        out[0] = light;
        out[1] = mel;
        out[2] = ema;
        out[3] = pe_s * (1.0f / 8.0f);
        out[4] = pr_s * (1.0f / 8.0f);
        out[5] = 0.5f * fe_s;
        out[6] = rt_s * (1.0f / 6.0f);
    }
#endif
}

extern "C" void kernel_launch(void* const* d_in, const int* in_sizes, int n_in,
                              void* d_out, int out_size, void* d_ws, size_t ws_size,
                              hipStream_t stream) {
    (void)in_sizes; (void)n_in; (void)out_size; (void)d_ws; (void)ws_size;
    const float* light_level       = (const float*)d_in[0];
    const float* retinal_luminance = (const float*)d_in[1];
    const float* melatonin_raw     = (const float*)d_in[2];
    const float* mel_ema           = (const float*)d_in[3];
    const float* v                 = (const float*)d_in[4];
    const float* u                 = (const float*)d_in[5];
    const float* rate              = (const float*)d_in[6];
    const float* noise             = (const float*)d_in[7];
    const float* v_soma            = (const float*)d_in[8];
    const float* v_apical          = (const float*)d_in[9];
    const float* precision         = (const float*)d_in[10];
    float* out = (float*)d_out;

    // Single wave32: the whole problem is launch-latency bound.
    SpikingPineal_kernel<<<1, 32, 0, stream>>>(
        light_level, retinal_luminance, melatonin_raw, mel_ema,
        v, u, rate, noise, v_soma, v_apical, precision, out);
}